// PGA_resnte_3435973837029
// MI455X (gfx1250) — compile-verified
//
#include <hip/hip_runtime.h>
#include <math.h>

// ---------------------------------------------------------------------------
// Problem constants (match reference)
// ---------------------------------------------------------------------------
#define BB   16
#define CC   256
#define HIDD 32
#define HH   64
#define WW2  32
#define NN   2048          // HH * WW2
#define BNEPS 1e-5f

typedef __attribute__((ext_vector_type(2))) float v2f;
typedef __attribute__((ext_vector_type(8))) float v8f;

// ---------------------------------------------------------------------------
// Kernel 1: alpha/sigma projections via f32 WMMA, fused BN + ReLU.
//   D[o, n] = sum_c W[o, c] * x[c, n]   (per batch), o in [0,32), n tile of 16
// One wave handles one 16-wide n-tile; 4 accumulators (wa lo/hi, ws lo/hi)
// share the same B fragment each K-step (K=4 per V_WMMA_F32_16X16X4_F32).
// Fragment layouts per ISA 7.12.2:
//   A 16x4 f32 : lanes 0-15 -> M rows, half-wave selects K pair {0,1}/{2,3}
//   B 4x16 f32 : lanes -> N cols, half-wave selects K pair
//   C/D 16x16  : VGPR v, lane L -> M = v + 8*(L>=16), N = L%16
// ---------------------------------------------------------------------------
__global__ __launch_bounds__(128)
void pga_gemm_bnrelu(const float* __restrict__ x,
                     const float* __restrict__ wa, const float* __restrict__ wsp,
                     const float* __restrict__ ga, const float* __restrict__ ba,
                     const float* __restrict__ ma, const float* __restrict__ va,
                     const float* __restrict__ gs, const float* __restrict__ bs,
                     const float* __restrict__ ms, const float* __restrict__ vs,
                     float* __restrict__ af, float* __restrict__ sf)
{
    const int lane  = threadIdx.x & 31;
    const int wave  = threadIdx.x >> 5;
    const int ntile = blockIdx.x * 4 + wave;   // 0..127
    const int b     = blockIdx.y;
    const int n0    = ntile * 16;
    const int half  = lane >> 4;               // 0: K{0,1}, 1: K{2,3}
    const int lm    = lane & 15;               // M (for A) / N (for B,C,D)

    const float* xb = x + (size_t)b * CC * NN;

    v8f acc_a0 = {0.f,0.f,0.f,0.f,0.f,0.f,0.f,0.f};
    v8f acc_a1 = {0.f,0.f,0.f,0.f,0.f,0.f,0.f,0.f};
    v8f acc_s0 = {0.f,0.f,0.f,0.f,0.f,0.f,0.f,0.f};
    v8f acc_s1 = {0.f,0.f,0.f,0.f,0.f,0.f,0.f,0.f};

    for (int k0 = 0; k0 < CC; k0 += 4) {
        const int ka = k0 + (half << 1);
        v2f bf;
        bf[0] = xb[(size_t)ka       * NN + n0 + lm];
        bf[1] = xb[(size_t)(ka + 1) * NN + n0 + lm];

        v2f a0, a1, s0, s1;
        a0[0] = wa [(size_t)lm        * CC + ka];
        a0[1] = wa [(size_t)lm        * CC + ka + 1];
        a1[0] = wa [(size_t)(lm + 16) * CC + ka];
        a1[1] = wa [(size_t)(lm + 16) * CC + ka + 1];
        s0[0] = wsp[(size_t)lm        * CC + ka];
        s0[1] = wsp[(size_t)lm        * CC + ka + 1];
        s1[0] = wsp[(size_t)(lm + 16) * CC + ka];
        s1[1] = wsp[(size_t)(lm + 16) * CC + ka + 1];

        acc_a0 = __builtin_amdgcn_wmma_f32_16x16x4_f32(false, a0, false, bf, (short)0, acc_a0, false, false);
        acc_a1 = __builtin_amdgcn_wmma_f32_16x16x4_f32(false, a1, false, bf, (short)0, acc_a1, false, false);
        acc_s0 = __builtin_amdgcn_wmma_f32_16x16x4_f32(false, s0, false, bf, (short)0, acc_s0, false, false);
        acc_s1 = __builtin_amdgcn_wmma_f32_16x16x4_f32(false, s1, false, bf, (short)0, acc_s1, false, false);
    }

    // BN (eval) + ReLU, then store af/sf as [B, HID, NN]
#pragma unroll
    for (int v = 0; v < 8; ++v) {
        const int o  = v + (half << 3);   // rows 0..15 of this tile
        const int o1 = o + 16;            // rows 16..31
        {
            float inv = ga[o] * rsqrtf(va[o] + BNEPS);
            float y   = acc_a0[v] * inv + (ba[o] - ma[o] * inv);
            af[((size_t)b * HIDD + o) * NN + n0 + lm] = fmaxf(y, 0.0f);
        }
        {
            float inv = ga[o1] * rsqrtf(va[o1] + BNEPS);
            float y   = acc_a1[v] * inv + (ba[o1] - ma[o1] * inv);
            af[((size_t)b * HIDD + o1) * NN + n0 + lm] = fmaxf(y, 0.0f);
        }
        {
            float inv = gs[o] * rsqrtf(vs[o] + BNEPS);
            float y   = acc_s0[v] * inv + (bs[o] - ms[o] * inv);
            sf[((size_t)b * HIDD + o) * NN + n0 + lm] = fmaxf(y, 0.0f);
        }
        {
            float inv = gs[o1] * rsqrtf(vs[o1] + BNEPS);
            float y   = acc_s1[v] * inv + (bs[o1] - ms[o1] * inv);
            sf[((size_t)b * HIDD + o1) * NN + n0 + lm] = fmaxf(y, 0.0f);
        }
    }
}

// ---------------------------------------------------------------------------
// Kernel 2: adjacent-channel dot products  d[b, i] = x[b,i,:] . x[b,i+1,:]
// (the only entries of Gc that survive the |i-j|==1 chain mask)
// ---------------------------------------------------------------------------
__global__ __launch_bounds__(256)
void pga_chandot(const float* __restrict__ x, float* __restrict__ d)
{
    const int i = blockIdx.x;      // 0..CC-2
    const int b = blockIdx.y;
    const float* r0 = x + ((size_t)b * CC + i) * NN;
    const float* r1 = r0 + NN;

    float s = 0.f;
    for (int j = threadIdx.x; j < NN; j += 256)
        s += r0[j] * r1[j];

    __shared__ float red[256];
    red[threadIdx.x] = s;
    __syncthreads();
#pragma unroll
    for (int st = 128; st > 0; st >>= 1) {
        if (threadIdx.x < st) red[threadIdx.x] += red[threadIdx.x + st];
        __syncthreads();
    }
    if (threadIdx.x == 0) d[b * CC + i] = red[0];
}

// ---------------------------------------------------------------------------
// Kernel 3: fused sparse spatial attention + chain channel attention +
//           ELU + residual blend. One thread = one pixel; channels tiled by 64
//           with a rolling x window; chain-softmax weights staged in LDS.
// ---------------------------------------------------------------------------
__global__ __launch_bounds__(128)
void pga_fuse(const float* __restrict__ x,
              const float* __restrict__ af, const float* __restrict__ sf,
              const float* __restrict__ d,
              const float* __restrict__ gama,
              float* __restrict__ out)
{
    const int b      = blockIdx.z;
    const int cstart = blockIdx.y * 64;
    const int j      = blockIdx.x * 128 + threadIdx.x;   // pixel id, < NN
    const int r      = j / WW2;
    const int col    = j - r * WW2;

    // chain-softmax channel weights for this channel tile
    __shared__ float s_wp[64], s_wn[64];
    if (threadIdx.x < 64) {
        const int c = cstart + threadIdx.x;
        float wp, wn;
        if (c == 0)            { wp = 0.f; wn = 1.f; }
        else if (c == CC - 1)  { wp = 1.f; wn = 0.f; }
        else {
            float ep = d[b * CC + c - 1];      // logit toward channel c-1
            float en = d[b * CC + c];          // logit toward channel c+1
            float m  = fmaxf(ep, en);
            float e1 = __expf(ep - m), e2 = __expf(en - m);
            float iz = 1.f / (e1 + e2);
            wp = e1 * iz; wn = e2 * iz;
        }
        s_wp[threadIdx.x] = wp;
        s_wn[threadIdx.x] = wn;
    }
    __syncthreads();

    // 8-connected neighbors of this pixel
    int nbr[8]; int cnt = 0;
#pragma unroll
    for (int dr = -1; dr <= 1; ++dr)
#pragma unroll
        for (int dc = -1; dc <= 1; ++dc) {
            if (dr == 0 && dc == 0) continue;
            int rr = r + dr, cc = col + dc;
            if (rr >= 0 && rr < HH && cc >= 0 && cc < WW2)
                nbr[cnt++] = rr * WW2 + cc;
        }

    // attention logits: l_k = sum_h af[h, j] * sf[h, nbr_k]
    const float* afb = af + (size_t)b * HIDD * NN;
    const float* sfb = sf + (size_t)b * HIDD * NN;
    float areg[HIDD];
#pragma unroll
    for (int h = 0; h < HIDD; ++h) areg[h] = afb[(size_t)h * NN + j];

    float wsm[8];
    float mx = -3.4e38f;
    for (int k = 0; k < cnt; ++k) {
        float s = 0.f;
#pragma unroll
        for (int h = 0; h < HIDD; ++h) s += areg[h] * sfb[(size_t)h * NN + nbr[k]];
        wsm[k] = s;
        mx = fmaxf(mx, s);
    }
    float z = 0.f;
    for (int k = 0; k < cnt; ++k) { float e = __expf(wsm[k] - mx); wsm[k] = e; z += e; }
    const float invz = 1.f / z;
    for (int k = 0; k < cnt; ++k) wsm[k] *= invz;

    const float g  = gama[0];
    const float* xb = x   + (size_t)b * CC * NN;
    float*       ob = out + (size_t)b * CC * NN;

    // rolling window over channels: xprev / xcur / xnext
    float xprev = (cstart > 0) ? xb[(size_t)(cstart - 1) * NN + j] : 0.f;
    float xcur  = xb[(size_t)cstart * NN + j];
    for (int ci = 0; ci < 64; ++ci) {
        const int c = cstart + ci;
        float xnext = (c + 1 < CC) ? xb[(size_t)(c + 1) * NN + j] : 0.f;

        float hs = 0.f;
        for (int k = 0; k < cnt; ++k)
            hs += wsm[k] * xb[(size_t)c * NN + nbr[k]];

        float hp = s_wp[ci] * xprev + s_wn[ci] * xnext;
        float hv = hs + hp;
        float he = (hv > 0.f) ? hv : (__expf(hv) - 1.f);   // ELU(alpha=1)

        ob[(size_t)c * NN + j] = (1.f - g) * xcur + g * he;

        xprev = xcur;
        xcur  = xnext;
    }
}

// ---------------------------------------------------------------------------
// Launch
// ---------------------------------------------------------------------------
extern "C" void kernel_launch(void* const* d_in, const int* in_sizes, int n_in,
                              void* d_out, int out_size, void* d_ws, size_t ws_size,
                              hipStream_t stream)
{
    (void)in_sizes; (void)n_in; (void)out_size; (void)ws_size;

    const float* x    = (const float*)d_in[0];
    const float* wa   = (const float*)d_in[1];
    const float* ga   = (const float*)d_in[2];
    const float* ba   = (const float*)d_in[3];
    const float* ma   = (const float*)d_in[4];
    const float* va   = (const float*)d_in[5];
    const float* wsp  = (const float*)d_in[6];   // 'ws' weights (name avoids clash)
    const float* gs   = (const float*)d_in[7];
    const float* bs   = (const float*)d_in[8];
    const float* ms   = (const float*)d_in[9];
    const float* vs   = (const float*)d_in[10];
    const float* gama = (const float*)d_in[11];
    float* out = (float*)d_out;

    char* wsb = (char*)d_ws;
    float* af = (float*)wsb;                                        // B*HID*NN f32 (4 MB)
    float* sf = (float*)(wsb + (size_t)BB * HIDD * NN * 4);          // 4 MB
    float* dd = (float*)(wsb + (size_t)2 * BB * HIDD * NN * 4);      // B*CC f32 (16 KB)

    dim3 g1(NN / 16 / 4, BB);        // (32, 16), 4 waves/block, 1 n-tile/wave
    pga_gemm_bnrelu<<<g1, 128, 0, stream>>>(x, wa, wsp, ga, ba, ma, va,
                                            gs, bs, ms, vs, af, sf);

    dim3 g2(CC - 1, BB);             // (255, 16)
    pga_chandot<<<g2, 256, 0, stream>>>(x, dd);

    dim3 g3(NN / 128, CC / 64, BB);  // (16, 4, 16) = 1024 blocks
    pga_fuse<<<g3, 128, 0, stream>>>(x, af, sf, dd, gama, out);
}